// AttentionLayer_13426067768107
// MI455X (gfx1250) — compile-verified
//
#include <hip/hip_runtime.h>
#include <hip/hip_bf16.h>
#include <math.h>

#define EMBED 128

typedef __attribute__((ext_vector_type(2))) float v2f;
typedef __attribute__((ext_vector_type(8))) float v8f;

// ---------------------------------------------------------------------------
// Kernel 0: zero the output (harness poisons d_out with 0xAA)
// ---------------------------------------------------------------------------
__global__ void zero_kernel(float* __restrict__ p, int n) {
    int i = blockIdx.x * blockDim.x + threadIdx.x;
    if (i < n) p[i] = 0.0f;
}

// ---------------------------------------------------------------------------
// Kernel 1: Q/K/V projections with V_WMMA_F32_16X16X4_F32 (exact fp32 path).
// One wave per (16-row tile, matrix-in-{Q,K,V}). A tile (16x128) kept in 64
// VGPRs; loop 8 N-tiles x 32 K-chunks -> 256 WMMAs per wave.
//   A layout (16x4 f32): lane<16 -> M=lane, K = kc*4 + {0,1}
//                        lane>=16 -> M=lane-16, K = kc*4 + {2,3}
//   B layout (4x16 f32): mirrored; B[k][n] = W[n][k]  (y = x @ W^T)
//   D layout (16x16 f32): VGPR g -> row g + 8*(lane>=16), col = lane&15
// ---------------------------------------------------------------------------
__global__ __launch_bounds__(256) void qkv_wmma_kernel(
    const float* __restrict__ E,
    const float* __restrict__ Wq, const float* __restrict__ Wk,
    const float* __restrict__ Wv,
    float* __restrict__ Q, float* __restrict__ Km, float* __restrict__ V,
    int n_nodes, int n_work)
{
    const int lane  = threadIdx.x & 31;
    const int wave  = threadIdx.x >> 5;
    const int work  = blockIdx.x * 8 + wave;
    if (work >= n_work) return;

    const int mat   = work % 3;
    const int mtile = work / 3;
    const int khalf = lane >> 4;          // 0 or 1
    const int lrow  = lane & 15;

    const float* W = (mat == 0) ? Wq : (mat == 1) ? Wk : Wv;
    float*       D = (mat == 0) ? Q  : (mat == 1) ? Km : V;

    int row_a = mtile * 16 + lrow;
    int row_c = (row_a < n_nodes) ? row_a : (n_nodes - 1);   // clamp (EXEC must stay full)

    // Load full 16x128 A tile into registers: 32 chunks of K=4
    v2f a[32];
    const float* arow = E + (size_t)row_c * EMBED + khalf * 2;
#pragma unroll
    for (int kc = 0; kc < 32; ++kc)
        a[kc] = *(const v2f*)(arow + kc * 4);

#pragma unroll 1
    for (int nt = 0; nt < 8; ++nt) {
        const int col = nt * 16 + lrow;
        const float* brow = W + (size_t)col * EMBED + khalf * 2;
        v8f acc = {};
#pragma unroll
        for (int kc = 0; kc < 32; ++kc) {
            v2f b = *(const v2f*)(brow + kc * 4);
            acc = __builtin_amdgcn_wmma_f32_16x16x4_f32(
                      false, a[kc], false, b, (short)0, acc, false, false);
        }
        const int rbase = mtile * 16 + 8 * khalf;
#pragma unroll
        for (int g = 0; g < 8; ++g) {
            int r = rbase + g;
            if (r < n_nodes)
                D[(size_t)r * EMBED + col] = acc[g];
        }
    }
}

// ---------------------------------------------------------------------------
// Kernel 2: per-edge scores  s[e] = dot(Q[src], K[dst]) / sqrt(128)
// One wave per edge: 32 lanes x float4 coalesced row loads, shuffle reduce.
// Also emits a per-block max for the global softmax.
// ---------------------------------------------------------------------------
__global__ __launch_bounds__(256) void score_kernel(
    const float* __restrict__ Q, const float* __restrict__ Km,
    const int* __restrict__ src, const int* __restrict__ dst,
    float* __restrict__ scores, float* __restrict__ blockMax, int n_edges)
{
    __shared__ float smax[8];
    const int lane = threadIdx.x & 31;
    const int wave = threadIdx.x >> 5;
    const int e    = blockIdx.x * 8 + wave;

    float s = -INFINITY;
    if (e < n_edges) {
        const int si = src[e], di = dst[e];
        const float4 q = *(const float4*)(Q  + (size_t)si * EMBED + lane * 4);
        const float4 k = *(const float4*)(Km + (size_t)di * EMBED + lane * 4);
        float d = q.x * k.x + q.y * k.y + q.z * k.z + q.w * k.w;
#pragma unroll
        for (int off = 16; off > 0; off >>= 1)
            d += __shfl_xor(d, off, 32);
        s = d * 0.088388347648318447f;   // 1/sqrt(128)
        if (lane == 0) scores[e] = s;
    }
    if (lane == 0) smax[wave] = s;
    __syncthreads();
    if (threadIdx.x == 0) {
        float m = smax[0];
#pragma unroll
        for (int i = 1; i < 8; ++i) m = fmaxf(m, smax[i]);
        blockMax[blockIdx.x] = m;
    }
}

// ---------------------------------------------------------------------------
// Kernel 3: global max of block maxima (single block)
// ---------------------------------------------------------------------------
__global__ __launch_bounds__(1024) void max_reduce_kernel(
    const float* __restrict__ in, int n, float* __restrict__ gmax)
{
    __shared__ float sm[1024];
    float m = -INFINITY;
    for (int i = threadIdx.x; i < n; i += 1024) m = fmaxf(m, in[i]);
    sm[threadIdx.x] = m;
    __syncthreads();
    for (int s = 512; s > 0; s >>= 1) {
        if ((int)threadIdx.x < s)
            sm[threadIdx.x] = fmaxf(sm[threadIdx.x], sm[threadIdx.x + s]);
        __syncthreads();
    }
    if (threadIdx.x == 0) gmax[0] = sm[0];
}

// ---------------------------------------------------------------------------
// Kernel 4: partial sums of exp(score - gmax)
// ---------------------------------------------------------------------------
__global__ __launch_bounds__(256) void expsum_kernel(
    const float* __restrict__ scores, int n_edges,
    const float* __restrict__ gmax, float* __restrict__ blockSum)
{
    __shared__ float ss[256];
    const float m = gmax[0];
    float acc = 0.0f;
    for (int i = blockIdx.x * 256 + threadIdx.x; i < n_edges; i += 256 * gridDim.x)
        acc += __expf(scores[i] - m);
    ss[threadIdx.x] = acc;
    __syncthreads();
    for (int s = 128; s > 0; s >>= 1) {
        if ((int)threadIdx.x < s) ss[threadIdx.x] += ss[threadIdx.x + s];
        __syncthreads();
    }
    if (threadIdx.x == 0) blockSum[blockIdx.x] = ss[0];
}

// ---------------------------------------------------------------------------
// Kernel 5: final sum (single block of 256 over 1024 partials)
// ---------------------------------------------------------------------------
__global__ __launch_bounds__(256) void sum_reduce_kernel(
    const float* __restrict__ in, int n, float* __restrict__ gsum)
{
    __shared__ float ss[256];
    float acc = 0.0f;
    for (int i = threadIdx.x; i < n; i += 256) acc += in[i];
    ss[threadIdx.x] = acc;
    __syncthreads();
    for (int s = 128; s > 0; s >>= 1) {
        if ((int)threadIdx.x < s) ss[threadIdx.x] += ss[threadIdx.x + s];
        __syncthreads();
    }
    if (threadIdx.x == 0) gsum[0] = ss[0];
}

// ---------------------------------------------------------------------------
// Kernel 6: weighted scatter  out[src] += V[dst] * softmax_weight(e)
// One wave per edge; 4 f32 atomics per lane (global_atomic_add_f32, L2-resident)
// ---------------------------------------------------------------------------
__global__ __launch_bounds__(256) void scatter_kernel(
    const float* __restrict__ V, const float* __restrict__ scores,
    const int* __restrict__ src, const int* __restrict__ dst,
    const float* __restrict__ gmax, const float* __restrict__ gsum,
    float* __restrict__ out, int n_edges)
{
    const int lane = threadIdx.x & 31;
    const int wave = threadIdx.x >> 5;
    const int e    = blockIdx.x * 8 + wave;
    if (e >= n_edges) return;

    const float w  = __expf(scores[e] - gmax[0]) / gsum[0];
    const int  si  = src[e], di = dst[e];
    const float4 v = *(const float4*)(V + (size_t)di * EMBED + lane * 4);
    float* o = out + (size_t)si * EMBED + lane * 4;
    atomicAdd(o + 0, v.x * w);
    atomicAdd(o + 1, v.y * w);
    atomicAdd(o + 2, v.z * w);
    atomicAdd(o + 3, v.w * w);
}

// ---------------------------------------------------------------------------
extern "C" void kernel_launch(void* const* d_in, const int* in_sizes, int n_in,
                              void* d_out, int out_size, void* d_ws, size_t ws_size,
                              hipStream_t stream) {
    const float* E   = (const float*)d_in[0];
    const int*   edg = (const int*)  d_in[1];
    const float* Wq  = (const float*)d_in[2];
    const float* Wk  = (const float*)d_in[3];
    const float* Wv  = (const float*)d_in[4];
    float*       out = (float*)d_out;

    const int n_nodes = in_sizes[0] / EMBED;
    const int n_edges = in_sizes[1] / 2;
    const int* src = edg;             // edge_index[0, :]
    const int* dst = edg + n_edges;   // edge_index[1, :]

    // Workspace layout (fp32): Q | K | V | scores | blockMax | gmax | blockSum | gsum
    const size_t nd  = (size_t)n_nodes * EMBED;
    float* Q        = (float*)d_ws;
    float* Km       = Q  + nd;
    float* V        = Km + nd;
    float* scores   = V  + nd;
    const int nbE   = (n_edges + 7) / 8;           // 8 edges (waves) per block
    float* blockMax = scores + n_edges;
    float* gmax     = blockMax + nbE;
    const int NB4   = 1024;
    float* blockSum = gmax + 1;
    float* gsum     = blockSum + NB4;

    // 0) zero output
    zero_kernel<<<(int)((nd + 255) / 256), 256, 0, stream>>>(out, (int)nd);

    // 1) Q/K/V GEMMs via fp32 WMMA
    const int n_work = ((n_nodes + 15) / 16) * 3;
    qkv_wmma_kernel<<<(n_work + 7) / 8, 256, 0, stream>>>(
        E, Wq, Wk, Wv, Q, Km, V, n_nodes, n_work);

    // 2) edge scores + per-block max
    score_kernel<<<nbE, 256, 0, stream>>>(Q, Km, src, dst, scores, blockMax, n_edges);

    // 3) global max
    max_reduce_kernel<<<1, 1024, 0, stream>>>(blockMax, nbE, gmax);

    // 4) exp-sum partials
    expsum_kernel<<<NB4, 256, 0, stream>>>(scores, n_edges, gmax, blockSum);

    // 5) final sum
    sum_reduce_kernel<<<1, 256, 0, stream>>>(blockSum, NB4, gsum);

    // 6) weighted atomic scatter
    scatter_kernel<<<nbE, 256, 0, stream>>>(V, scores, src, dst, gmax, gsum, out, n_edges);
}